// DistanceCentroidLoss_48223892800209
// MI455X (gfx1250) — compile-verified
//
#include <hip/hip_runtime.h>

typedef float v2f __attribute__((ext_vector_type(2)));
typedef float v4f __attribute__((ext_vector_type(4)));
typedef float v8f __attribute__((ext_vector_type(8)));

static constexpr int kN  = 65536;   // points
static constexpr int kD  = 512;     // dims
static constexpr int kK  = 256;     // clusters
static constexpr int TM  = 64;      // rows per workgroup tile
static constexpr int DC  = 64;      // D-chunk staged in LDS
static constexpr int LDA = 68;      // sA row stride (floats), conflict-free frag reads
static constexpr int LDB = 257;     // sB row stride (float2 pairs), conflict-free
static constexpr int LSQ = 257;     // sSq row stride (floats), conflict-free column scans
static constexpr float kMargin = 10.0f;

// workspace layout (floats): [0,256) cnorm  [256,512) attr  [512,768) rep  [768,1024) cnt

// ---------------------------------------------------------------- init -----
__global__ void dcl_init(const float* __restrict__ cent, float* __restrict__ ws) {
    __shared__ float red[64];
    const int k = blockIdx.x;      // one block per centroid
    const int t = threadIdx.x;     // 64 threads
    const float* row = cent + (size_t)k * kD;
    float s = 0.f;
    for (int j = t; j < kD; j += 64) { float v = row[j]; s += v * v; }
    red[t] = s;
    __syncthreads();
    for (int off = 32; off > 0; off >>= 1) {
        if (t < off) red[t] += red[t + off];
        __syncthreads();
    }
    if (t == 0) {
        ws[k]       = red[0];  // ||c_k||^2
        ws[256 + k] = 0.f;     // attraction accum
        ws[512 + k] = 0.f;     // repulsion accum
        ws[768 + k] = 0.f;     // count accum
    }
}

// ---------------------------------------------------------------- main -----
__global__ __launch_bounds__(256) void
dcl_main(const float* __restrict__ emb, const int* __restrict__ labels,
         const float* __restrict__ cent, float* __restrict__ ws) {
    __shared__ __align__(16) float sA[TM * LDA];          // 17408 B
    __shared__ __align__(16) v2f   sB[(DC / 2) * LDB];    // 65792 B (reused as sSq)
    __shared__ float sRow[TM];                            // ||p||^2 per tile row
    __shared__ float sCn[kK];                             // ||c||^2 cache
    __shared__ float sAttr[kK], sRep[kK], sCnt[kK];       // per-cluster bins

    const int t     = threadIdx.x;
    const int lane  = t & 31;
    const int wid   = t >> 5;
    const int strip = wid & 3;        // 16-row strip within tile
    const int chalf = wid >> 2;       // column half (0..127 / 128..255)
    const int h     = lane >> 4;      // lane half (K/M offset selector)
    const int ml    = lane & 15;      // m (A) / n (B) within 16
    const int r0    = blockIdx.x * TM;

    // stage-independent init (ordered by the first barrier below)
    if (t < TM) sRow[t] = 0.f;
    sCn[t]   = ws[t];
    sAttr[t] = 0.f; sRep[t] = 0.f; sCnt[t] = 0.f;

    v8f acc[8];
    const v8f vzero = {0.f, 0.f, 0.f, 0.f, 0.f, 0.f, 0.f, 0.f};
#pragma unroll
    for (int i = 0; i < 8; ++i) acc[i] = vzero;

    const int arow_lo = t >> 4;   // this thread feeds rows {arow_lo + 16*i}
    const int ac4     = t & 15;   // float4 column group
    float ssq[4] = {0.f, 0.f, 0.f, 0.f};

    for (int d0 = 0; d0 < kD; d0 += DC) {
        // ---- stage A chunk (64 x 64) + accumulate row sum-of-squares ----
#pragma unroll
        for (int i = 0; i < 4; ++i) {
            const int row = 16 * i + arow_lo;
            const v4f v = *(const v4f*)(emb + (size_t)(r0 + row) * kD + d0 + ac4 * 4);
            ssq[i] += v.x * v.x + v.y * v.y + v.z * v.z + v.w * v.w;
            float* dst = &sA[row * LDA + ac4 * 4];
            *(v2f*)(dst)     = (v2f){v.x, v.y};
            *(v2f*)(dst + 2) = (v2f){v.z, v.w};
        }
        // ---- stage B chunk as K-pairs: sB[kp*LDB + n] = (C[n][d0+2kp], C[n][d0+2kp+1])
#pragma unroll 4
        for (int i = 0; i < 32; ++i) {
            const int idx = t + 256 * i;
            const int n   = idx >> 5;
            const int kp  = idx & 31;
            sB[kp * LDB + n] = *(const v2f*)(cent + (size_t)n * kD + d0 + 2 * kp);
        }
        if (d0 + DC < kD)  // hint next A chunk toward the caches
            __builtin_prefetch(emb + (size_t)(r0 + arow_lo) * kD + d0 + DC, 0, 0);
        __syncthreads();

        // ---- WMMA core: 16 K-steps x 8 column tiles ----
#pragma unroll
        for (int k = 0; k < DC; k += 4) {
            const v2f a = *(const v2f*)&sA[(strip * 16 + ml) * LDA + k + 2 * h];
#pragma unroll
            for (int nt = 0; nt < 8; ++nt) {
                const v2f b = sB[(k / 2 + h) * LDB + chalf * 128 + nt * 16 + ml];
                acc[nt] = __builtin_amdgcn_wmma_f32_16x16x4_f32(
                    false, a, false, b, (short)0, acc[nt], false, false);
            }
        }
        __syncthreads();
    }

    // ---- spill dot tile to LDS (overlay on sB) ----
    float* sSq = (float*)sB;   // 64 * 257 floats fits in sB
#pragma unroll
    for (int nt = 0; nt < 8; ++nt) {
        const int col = chalf * 128 + nt * 16 + ml;
#pragma unroll
        for (int v = 0; v < 8; ++v)
            sSq[(strip * 16 + v + 8 * h) * LSQ + col] = acc[nt][v];
    }
#pragma unroll
    for (int i = 0; i < 4; ++i) atomicAdd(&sRow[16 * i + arow_lo], ssq[i]);
    __syncthreads();

    // ---- fused per-point pass: own distance + top-8 smallest others ----
    if (t < TM) {
        const int   gi  = r0 + t;
        const int   lbl = labels[gi];
        const float pn  = sRow[t];
        const float* dotrow = sSq + t * LSQ;
        float own = 0.f;
        float best[8];
#pragma unroll
        for (int j = 0; j < 8; ++j) best[j] = 3.4e38f;

        for (int k = 0; k < kK; ++k) {
            float sq = fmaxf(pn + sCn[k] - 2.0f * dotrow[k], 0.f);
            if (k == lbl) { own = sq; continue; }
            if (sq < best[7]) {
                best[7] = sq;
#pragma unroll
                for (int j = 7; j > 0; --j) {
                    if (best[j] < best[j - 1]) {
                        float tmp = best[j]; best[j] = best[j - 1]; best[j - 1] = tmp;
                    }
                }
            }
        }
        float pp = 0.f;
#pragma unroll
        for (int j = 0; j < 8; ++j) {
            const float d = sqrtf(best[j] + 1e-12f);
            const float m = kMargin - d;
            pp += m * m;
        }
        pp *= 0.125f;
        atomicAdd(&sAttr[lbl], own);
        atomicAdd(&sRep[lbl],  pp);
        atomicAdd(&sCnt[lbl],  1.0f);
    }
    __syncthreads();

    // ---- one global atomic flush per cluster per WG ----
    atomicAdd(&ws[256 + t], sAttr[t]);
    atomicAdd(&ws[512 + t], sRep[t]);
    atomicAdd(&ws[768 + t], sCnt[t]);
}

// -------------------------------------------------------------- finalize ---
__global__ void dcl_final(const float* __restrict__ ws, float* __restrict__ out) {
    __shared__ float red[256];
    const int t = threadIdx.x;
    const float c = fmaxf(ws[768 + t], 1.0f);
    red[t] = (ws[256 + t] + ws[512 + t]) / c;
    __syncthreads();
    for (int off = 128; off > 0; off >>= 1) {
        if (t < off) red[t] += red[t + off];
        __syncthreads();
    }
    if (t == 0) out[0] = red[0] / (float)kK;
}

// ---------------------------------------------------------------- launch ---
extern "C" void kernel_launch(void* const* d_in, const int* in_sizes, int n_in,
                              void* d_out, int out_size, void* d_ws, size_t ws_size,
                              hipStream_t stream) {
    const float* emb    = (const float*)d_in[0];   // [N, D] f32
    const int*   labels = (const int*)d_in[1];     // [N] i32
    const float* cent   = (const float*)d_in[2];   // [K, D] f32
    float* ws  = (float*)d_ws;                     // 1024 floats used
    float* out = (float*)d_out;

    dcl_init <<<kK, 64, 0, stream>>>(cent, ws);
    dcl_main <<<kN / TM, 256, 0, stream>>>(emb, labels, cent, ws);
    dcl_final<<<1, 256, 0, stream>>>(ws, out);
}